// GAT_43611097924208
// MI455X (gfx1250) — compile-verified
//
#include <hip/hip_runtime.h>

// ---------------------------------------------------------------------------
// GAT (2x GATConv) for MI455X / gfx1250.
//  - GEMMs: V_WMMA_F32_16X16X4_F32, one wave per 16x16 tile, A tile staged in
//    LDS (padded stride -> conflict-free fragment reads), W streamed from L2.
//  - Edge phase (the real bottleneck: ~1 GB gather + ~220M f32 atomics):
//    3-pass segment softmax with L2-resident atomics; aggregation is
//    wave-per-edge with float4 gathers (coalesced 512B/edge).
//  - wave32-safe: EXEC all-ones through WMMA (clamped reads, guarded writes).
// ---------------------------------------------------------------------------

typedef float v2f __attribute__((ext_vector_type(2)));
typedef float v8f __attribute__((ext_vector_type(8)));

#define NEG_SLOPE 0.2f
#define LDA 132   // 128 + 4 pad: fragment reads hit banks 4*m (+k), all distinct

__device__ __forceinline__ int ord_of_f32(float f) {
  int i = __float_as_int(f);
  return (i >= 0) ? i : (i ^ 0x7FFFFFFF);   // monotonic float->int mapping
}
__device__ __forceinline__ float f32_of_ord(int i) {
  return __int_as_float((i >= 0) ? i : (i ^ 0x7FFFFFFF));
}

// C[M,128] = A[M,128] @ W[128,128]; fp32 WMMA.
// grid = (ceil(M/16), 2), block = 128 (4 waves; wave w owns 16-col tile).
__global__ void __launch_bounds__(128)
gat_gemm_wmma_f32(const float* __restrict__ A, const float* __restrict__ W,
                  float* __restrict__ C, int M) {
  constexpr int K = 128, Nc = 128;
  __shared__ float sA[16 * LDA];

  const int lane = threadIdx.x & 31;
  const int wave = threadIdx.x >> 5;
  const int row0 = blockIdx.x * 16;

  // Cooperative stage of the 16x128 A tile (shared by all 4 waves).
  {
    const int t = threadIdx.x;            // 0..127
    const int r = t >> 3;                 // 16 rows, 8 threads per row
    const int cb = (t & 7) << 4;          // 16 floats per thread
    int rr = row0 + r;
    rr = (rr < M) ? rr : (M - 1);         // clamp: keep loads in-bounds
    const float4* src = (const float4*)(A + (size_t)rr * K + cb);
    float4 v0 = src[0], v1 = src[1], v2 = src[2], v3 = src[3];
    float* dp = sA + r * LDA + cb;
    *(float4*)(dp + 0)  = v0;
    *(float4*)(dp + 4)  = v1;
    *(float4*)(dp + 8)  = v2;
    *(float4*)(dp + 12) = v3;
  }
  __syncthreads();

  const int col0 = (blockIdx.y * 4 + wave) * 16;
  const int n  = col0 + (lane & 15);
  const int kh = (lane >> 4) << 1;                 // 0 | 2
  const float* __restrict__ ar = sA + (lane & 15) * LDA;

  v8f acc = {};
  for (int kk = 0; kk < K; kk += 4) {
    v2f a, b;
    a.x = ar[kk + kh];
    a.y = ar[kk + kh + 1];
    b.x = W[(size_t)(kk + kh)     * Nc + n];
    b.y = W[(size_t)(kk + kh + 1) * Nc + n];
    acc = __builtin_amdgcn_wmma_f32_16x16x4_f32(
        /*neg_a=*/false, a, /*neg_b=*/false, b,
        /*c_mod=*/(short)0, acc, /*reuse_a=*/false, /*reuse_b=*/false);
  }

  // C/D layout: VGPR r -> M = r (lanes 0-15) or r+8 (lanes 16-31), N = lane&15
  const int mBase = row0 + ((lane >> 4) << 3);
#pragma unroll
  for (int r = 0; r < 8; ++r) {
    int m = mBase + r;
    if (m < M) C[(size_t)m * Nc + n] = acc[r];
  }
}

// Per (node, head): alpha_src/alpha_dst dots; also init segment-max/denominator.
__global__ void node_alpha_init(const float* __restrict__ h,
                                const float* __restrict__ att_src,
                                const float* __restrict__ att_dst,
                                float* __restrict__ aS, float* __restrict__ aD,
                                int* __restrict__ emax, float* __restrict__ den,
                                int N, int hShift, int C) {
  int t = blockIdx.x * blockDim.x + threadIdx.x;
  int H = 1 << hShift;
  if (t >= N * H) return;
  int nd = t >> hShift;
  int hd = t & (H - 1);
  const float4* hp = (const float4*)(h + (size_t)nd * (H * C) + hd * C);
  const float4* as = (const float4*)(att_src + hd * C);
  const float4* ad = (const float4*)(att_dst + hd * C);
  float s = 0.f, d = 0.f;
  for (int c = 0; c < (C >> 2); ++c) {
    float4 v = hp[c], a = as[c], b = ad[c];
    s = fmaf(v.x, a.x, fmaf(v.y, a.y, fmaf(v.z, a.z, fmaf(v.w, a.w, s))));
    d = fmaf(v.x, b.x, fmaf(v.y, b.y, fmaf(v.z, b.z, fmaf(v.w, b.w, d))));
  }
  aS[t] = s;
  aD[t] = d;
  emax[t] = (int)0x80000000;   // ordered(-inf); every node has a self-loop
  den[t]  = 0.f;
}

// out[n, :] = bias  (aggregation then atomically accumulates on top)
__global__ void init_out_bias(float* __restrict__ out, const float* __restrict__ b,
                              int total, int HC) {
  int t = blockIdx.x * blockDim.x + threadIdx.x;
  if (t >= total) return;
  out[t] = b[t & (HC - 1)];
}

// Pass 1: per-dst segment max of leaky_relu(aS[src]+aD[dst]).
__global__ void edge_max(const int* __restrict__ ei, int E, int N,
                         const float* __restrict__ aS, const float* __restrict__ aD,
                         int* __restrict__ emax, int hShift) {
  int t = blockIdx.x * blockDim.x + threadIdx.x;
  int H = 1 << hShift;
  int total = (E + N) * H;
  if (t >= total) return;
  int eid = t >> hShift;
  int hd  = t & (H - 1);
  int s, d;
  if (eid < E) { s = ei[eid]; d = ei[E + eid]; } else { s = d = eid - E; }
  float e = aS[(s << hShift) + hd] + aD[(d << hShift) + hd];
  e = (e > 0.f) ? e : NEG_SLOPE * e;
  atomicMax(&emax[(d << hShift) + hd], ord_of_f32(e));
}

// Pass 2: ex = exp(e - emax[dst]); denom[dst] += ex; stash ex per edge.
__global__ void edge_expsum(const int* __restrict__ ei, int E, int N,
                            const float* __restrict__ aS, const float* __restrict__ aD,
                            const int* __restrict__ emax, float* __restrict__ ex,
                            float* __restrict__ den, int hShift) {
  int t = blockIdx.x * blockDim.x + threadIdx.x;
  int H = 1 << hShift;
  int total = (E + N) * H;
  if (t >= total) return;
  int eid = t >> hShift;
  int hd  = t & (H - 1);
  int s, d;
  if (eid < E) { s = ei[eid]; d = ei[E + eid]; } else { s = d = eid - E; }
  float e = aS[(s << hShift) + hd] + aD[(d << hShift) + hd];
  e = (e > 0.f) ? e : NEG_SLOPE * e;
  float m = f32_of_ord(emax[(d << hShift) + hd]);
  float x = __expf(e - m);
  ex[t] = x;
  atomicAdd(&den[(d << hShift) + hd], x);
}

// Pass 3: out[dst, :] += (ex/denom[dst]) * h[src, :]. One WAVE per edge:
// eid is wave-uniform (index loads scalarize), each lane owns 4 contiguous
// channels -> one b128 gather + 4 atomic f32 adds. HC = 128 fixed.
__global__ void __launch_bounds__(256)
edge_aggregate(const int* __restrict__ ei, int E, int N,
               const float* __restrict__ h, const float* __restrict__ ex,
               const float* __restrict__ den, float* __restrict__ out,
               int hShift, int cShift) {
  const int lane = threadIdx.x & 31;
  const int eid  = blockIdx.x * 8 + (threadIdx.x >> 5);   // 8 waves/block
  if (eid >= E + N) return;
  int s, d;
  if (eid < E) { s = ei[eid]; d = ei[E + eid]; } else { s = d = eid - E; }
  const int c0 = lane << 2;            // 4 channels per lane (same head: 4|C)
  const int hd = c0 >> cShift;
  float dn = den[(d << hShift) + hd];
  dn = (dn > 1e-16f) ? dn : 1e-16f;
  const float alpha = ex[((size_t)eid << hShift) + hd] / dn;
  const float4 hv = *(const float4*)(h + ((size_t)s << 7) + c0);
  float* op = out + ((size_t)d << 7) + c0;
  atomicAdd(op + 0, alpha * hv.x);
  atomicAdd(op + 1, alpha * hv.y);
  atomicAdd(op + 2, alpha * hv.z);
  atomicAdd(op + 3, alpha * hv.w);
}

__global__ void relu_inplace(float* __restrict__ x, int total) {
  int t = blockIdx.x * blockDim.x + threadIdx.x;
  if (t >= total) return;
  float v = x[t];
  x[t] = (v > 0.f) ? v : 0.f;
}

static inline unsigned cdiv(long long a, int b) { return (unsigned)((a + b - 1) / b); }

extern "C" void kernel_launch(void* const* d_in, const int* in_sizes, int n_in,
                              void* d_out, int out_size, void* d_ws, size_t ws_size,
                              hipStream_t stream) {
  const float* x   = (const float*)d_in[0];
  const int*   ei  = (const int*)d_in[1];   // edge_index [2,E], int32
  const float* W1  = (const float*)d_in[2];
  const float* as1 = (const float*)d_in[3];
  const float* ad1 = (const float*)d_in[4];
  const float* b1  = (const float*)d_in[5];
  const float* W2  = (const float*)d_in[6];
  const float* as2 = (const float*)d_in[7];
  const float* ad2 = (const float*)d_in[8];
  const float* b2  = (const float*)d_in[9];
  float* out = (float*)d_out;

  const int F = 128, HC = 128;
  const int H1s = 2, C1 = 32, C1s = 5;   // layer1: H=4, C=32
  const int H2s = 0, C2 = 128, C2s = 7;  // layer2: H=1, C=128
  const int N  = in_sizes[0] / F;
  const int E  = in_sizes[1] / 2;
  const int ET = E + N;                  // edges incl. self loops

  // Workspace carve-up (fp32 elements)
  float* p = (float*)d_ws;
  float* h1    = p; p += (size_t)N * HC;
  float* aS1   = p; p += (size_t)N * 4;
  float* aD1   = p; p += (size_t)N * 4;
  int*   emax1 = (int*)p; p += (size_t)N * 4;
  float* den1  = p; p += (size_t)N * 4;
  float* ex1   = p; p += (size_t)ET * 4;
  float* agg1  = p; p += (size_t)N * HC;
  float* h2    = p; p += (size_t)N * HC;
  float* aS2   = p; p += N;
  float* aD2   = p; p += N;
  int*   emax2 = (int*)p; p += N;
  float* den2  = p; p += N;
  float* ex2   = p; p += ET;

  const int TB = 256;
  dim3 gg(cdiv(N, 16), 2);

  // ---------------- Layer 1 (heads=4, out_ch=32) ----------------
  gat_gemm_wmma_f32<<<gg, 128, 0, stream>>>(x, W1, h1, N);
  node_alpha_init<<<cdiv((long long)N * 4, TB), TB, 0, stream>>>(
      h1, as1, ad1, aS1, aD1, emax1, den1, N, H1s, C1);
  init_out_bias<<<cdiv((long long)N * HC, TB), TB, 0, stream>>>(agg1, b1, N * HC, HC);
  edge_max<<<cdiv((long long)ET * 4, TB), TB, 0, stream>>>(ei, E, N, aS1, aD1, emax1, H1s);
  edge_expsum<<<cdiv((long long)ET * 4, TB), TB, 0, stream>>>(
      ei, E, N, aS1, aD1, emax1, ex1, den1, H1s);
  edge_aggregate<<<cdiv(ET, 8), TB, 0, stream>>>(
      ei, E, N, h1, ex1, den1, agg1, H1s, C1s);
  relu_inplace<<<cdiv((long long)N * HC, TB), TB, 0, stream>>>(agg1, N * HC);

  // ---------------- Layer 2 (heads=1, out_ch=128) ----------------
  gat_gemm_wmma_f32<<<gg, 128, 0, stream>>>(agg1, W2, h2, N);
  node_alpha_init<<<cdiv((long long)N, TB), TB, 0, stream>>>(
      h2, as2, ad2, aS2, aD2, emax2, den2, N, H2s, C2);
  init_out_bias<<<cdiv((long long)N * HC, TB), TB, 0, stream>>>(out, b2, N * HC, HC);
  edge_max<<<cdiv((long long)ET, TB), TB, 0, stream>>>(ei, E, N, aS2, aD2, emax2, H2s);
  edge_expsum<<<cdiv((long long)ET, TB), TB, 0, stream>>>(
      ei, E, N, aS2, aD2, emax2, ex2, den2, H2s);
  edge_aggregate<<<cdiv(ET, 8), TB, 0, stream>>>(
      ei, E, N, h2, ex2, den2, out, H2s, C2s);

  (void)n_in; (void)out_size; (void)ws_size;
}